// GumbelLinear_76106820485854
// MI455X (gfx1250) — compile-verified
//
#include <hip/hip_runtime.h>
#include <math.h>

// Reference: GumbelLinear — 64x16 input, 16x16 weights, top-5 hard mask output.
// Single-workgroup (128 threads = 4 wave32) kernel; GEMM done with
// V_WMMA_F32_16X16X4_F32 (one 16x16 output tile per wave, 4 chained K=4 steps).

typedef __attribute__((ext_vector_type(2))) float v2f;
typedef __attribute__((ext_vector_type(8))) float v8f;

#define NB 64
#define ND 16
#define EPSV 1e-8f

__global__ __launch_bounds__(128) void gumbel_linear_kernel(
    const float* __restrict__ h_in,
    const float* __restrict__ w_p,
    const float* __restrict__ bias,
    const float* __restrict__ U1,
    const float* __restrict__ U2,
    float* __restrict__ out)
{
    __shared__ float sh[NB * ND];     // remapped h
    __shared__ float ssoft[NB * ND];  // sigmoid outputs
    __shared__ float rmin[128];
    __shared__ float rmax[128];

    const int tid = threadIdx.x;          // 0..127

    // ---- 1) load h (coalesced, 8 elems/thread), local min/max ----
    float vals[8];
    float lmin =  INFINITY;
    float lmax = -INFINITY;
    #pragma unroll
    for (int i = 0; i < 8; ++i) {
        const float v = h_in[i * 128 + tid];
        vals[i] = v;
        lmin = fminf(lmin, v);
        lmax = fmaxf(lmax, v);
    }
    rmin[tid] = lmin;
    rmax[tid] = lmax;
    __syncthreads();

    // ---- 2) block tree reduction for global min/max ----
    #pragma unroll
    for (int s = 64; s > 0; s >>= 1) {
        if (tid < s) {
            rmin[tid] = fminf(rmin[tid], rmin[tid + s]);
            rmax[tid] = fmaxf(rmax[tid], rmax[tid + s]);
        }
        __syncthreads();
    }
    const float hmin = rmin[0];
    const float hmax = rmax[0];
    const bool  oor  = (hmax > 100.0f) || (hmin < -100.0f);
    const float scale = 0.6f / (hmax - hmin);

    // ---- 3) conditional remap, stage h into LDS ----
    #pragma unroll
    for (int i = 0; i < 8; ++i) {
        float v = vals[i];
        if (oor) {
            v = (v - hmin) * scale - 0.3f;
            v = fminf(fmaxf(v, -0.3f), 0.3f);
        }
        sh[i * 128 + tid] = v;
    }
    __syncthreads();

    // ---- 4) per-wave 16x16 tile of mask = h @ w_p via WMMA f32 16x16x4 ----
    const int wave = tid >> 5;      // 0..3 -> rows [16*wave, 16*wave+15]
    const int lane = tid & 31;
    const int half = lane >> 4;     // 0: lanes 0-15, 1: lanes 16-31
    const int l16  = lane & 15;
    const int r0   = wave * 16;

    v8f c = {};
    #pragma unroll
    for (int s = 0; s < 4; ++s) {
        const int k0 = 4 * s + 2 * half;
        v2f a, b;
        // A 16x4 tile: lane holds row M=l16, K = k0, k0+1
        a.x = sh[(r0 + l16) * ND + (k0 + 0)];
        a.y = sh[(r0 + l16) * ND + (k0 + 1)];
        // B 4x16 tile: lane holds col N=l16, K = k0, k0+1
        b.x = w_p[(k0 + 0) * ND + l16];
        b.y = w_p[(k0 + 1) * ND + l16];
        c = __builtin_amdgcn_wmma_f32_16x16x4_f32(
                /*neg_a=*/false, a, /*neg_b=*/false, b,
                /*c_mod=*/(short)0, c, /*reuse_a=*/false, /*reuse_b=*/false);
    }

    // ---- 5) fuse bias + gumbel + sigmoid; stage soft into LDS ----
    #pragma unroll
    for (int v = 0; v < 8; ++v) {
        const int row = r0 + v + 8 * half;   // C/D layout: VGPR v, half selects M or M+8
        const int idx = row * ND + l16;
        const float m  = c[v] + bias[idx];
        const float g1 = -logf(-logf(U1[idx] + EPSV) + EPSV);
        const float g2 = -logf(-logf(U2[idx] + EPSV) + EPSV);
        const float x  = m + g1 - g2;        // GUMBEL_TEMP == 1
        const float soft = 1.0f / (1.0f + expf(-x));
        ssoft[idx] = soft;
    }
    __syncthreads();

    // ---- 6) per-row top-5 threshold -> hard 0/1 mask ----
    if (tid < NB) {
        float v[ND];
        #pragma unroll
        for (int i = 0; i < ND; ++i) v[i] = ssoft[tid * ND + i];

        unsigned used = 0u;
        float fifth = -INFINITY;
        #pragma unroll
        for (int k = 0; k < 5; ++k) {
            float best = -INFINITY;
            int   bi   = 0;
            #pragma unroll
            for (int i = 0; i < ND; ++i) {
                const bool ok = ((used >> i) & 1u) == 0u;
                if (ok && v[i] > best) { best = v[i]; bi = i; }
            }
            used |= (1u << bi);
            fifth = best;
        }
        #pragma unroll
        for (int i = 0; i < ND; ++i) {
            out[tid * ND + i] = (v[i] >= fifth) ? 1.0f : 0.0f;
        }
    }
}

extern "C" void kernel_launch(void* const* d_in, const int* in_sizes, int n_in,
                              void* d_out, int out_size, void* d_ws, size_t ws_size,
                              hipStream_t stream) {
    (void)in_sizes; (void)n_in; (void)out_size; (void)d_ws; (void)ws_size;
    const float* h    = (const float*)d_in[0];
    // d_in[1] ("input") is unused by the reference computation (only shape).
    const float* w_p  = (const float*)d_in[2];
    const float* bias = (const float*)d_in[3];
    const float* U1   = (const float*)d_in[4];
    const float* U2   = (const float*)d_in[5];
    float* out = (float*)d_out;

    gumbel_linear_kernel<<<1, 128, 0, stream>>>(h, w_p, bias, U1, U2, out);
}